// CTCLoss_5205500363038
// MI455X (gfx1250) — compile-verified
//
#include <hip/hip_runtime.h>
#include <stdint.h>

// Problem constants (fixed by the reference)
#define Tn 512
#define Bn 64
#define Vn 512
#define Ln 128
#define Sn 257              // 2*Ln + 1
#define NEGV (-1e30f)

// The async builtin expects v4i pointers in explicit address spaces:
//   (v4i addrspace(1)* src, v4i addrspace(3)* dst, imm offset, imm cpol)
typedef int v4i __attribute__((ext_vector_type(4)));
typedef __attribute__((address_space(1))) v4i global_v4i;
typedef __attribute__((address_space(3))) v4i lds_v4i;

#if defined(__has_builtin)
#  if __has_builtin(__builtin_amdgcn_global_load_async_to_lds_b128)
#    define HAVE_ASYNC_LDS 1
#  endif
#  if __has_builtin(__builtin_amdgcn_s_wait_asynccnt)
#    define HAVE_WAIT_ASYNC 1
#  endif
#endif

// Async copy of 16 bytes (one float4) global -> LDS, tracked by ASYNCcnt.
__device__ __forceinline__ void async_cp_b128(const float* gsrc, float* ldst) {
#ifdef HAVE_ASYNC_LDS
  __builtin_amdgcn_global_load_async_to_lds_b128(
      (global_v4i*)(uintptr_t)gsrc,
      (lds_v4i*)(unsigned int)(uintptr_t)ldst,
      /*offset=*/0, /*cpol=*/0);
#else
#warning "global_load_async_to_lds_b128 builtin unavailable: synchronous LDS staging fallback"
  *(float4*)ldst = *(const float4*)gsrc;
#endif
}

template <int N>
__device__ __forceinline__ void wait_async_le() {
#ifdef HAVE_WAIT_ASYNC
  __builtin_amdgcn_s_wait_asynccnt(N);
#elif defined(HAVE_ASYNC_LDS)
  asm volatile("s_wait_asynccnt %0" ::"n"(N) : "memory");
#endif
}

// ---------------------------------------------------------------------------
// CTC forward chain. One block (288 thr = 9 waves) per batch element.
// The 2KB logits row for timestep t+3 is prefetched global->LDS with async
// b128 copies into a 3-slot ring (threads 0..127 issue one b128 each
// => 1 ASYNCcnt per wave per row; in-order completion => s_wait_asynccnt <= 2
// proves row t resident). Alpha is double-buffered in LDS.
// ---------------------------------------------------------------------------
__global__ __launch_bounds__(288) void ctc_fwd_kernel(const float* __restrict__ preds,
                                                      const int*   __restrict__ labels,
                                                      const int*   __restrict__ plens,
                                                      const int*   __restrict__ llens,
                                                      const float* __restrict__ lse,
                                                      float*       __restrict__ loss_out) {
  __shared__ float rowbuf[3][Vn];
  __shared__ float alpha[2][Sn];
  __shared__ int   exts[Sn];
  __shared__ int   allow[Sn];

  const int b   = blockIdx.x;
  const int tid = threadIdx.x;

  // Prime the ring FIRST: rows 0,1,2 in flight (1 ASYNCcnt per issuing wave
  // per row). Their DMA overlaps the label-table setup below.
  if (tid < 128) {
    const float* g = preds + (size_t)b * Vn + tid * 4;
    async_cp_b128(g,                     &rowbuf[0][tid * 4]);
    async_cp_b128(g + (size_t)Bn * Vn,   &rowbuf[1][tid * 4]);
    async_cp_b128(g + (size_t)2 * Bn * Vn, &rowbuf[2][tid * 4]);
  }

  const int plen = plens[b];
  const int llen = llens[b];

  // Extended label sequence: blank, l1, blank, l2, ..., blank.
  if (tid < Sn) {
    int e = 0, al = 0;
    if (tid & 1) {
      e = labels[b * Ln + (tid >> 1)];
      if (tid >= 3) al = (e != labels[b * Ln + ((tid - 3) >> 1)]) ? 1 : 0;
    }
    exts[tid]  = e;
    allow[tid] = al;
  }

  int cur = 0;
  int slot = 0;                        // = t % 3
  for (int t = 0; t < Tn; ++t) {
    // Row t complete for this wave (rows t+1, t+2 may still be in flight)…
    if (t + 2 < Tn)      wait_async_le<2>();
    else if (t + 1 < Tn) wait_async_le<1>();
    else                 wait_async_le<0>();
    __syncthreads();                   // …and for every wave in the block.

    const float lse_t = lse[b * Tn + t];

    if (tid < Sn) {
      const float e = rowbuf[slot][exts[tid]] - lse_t;   // logp gather from LDS
      float nv;
      if (t == 0) {
        nv = (tid == 0) ? e : ((tid == 1 && llen > 0) ? e : NEGV);
      } else {
        const float a0 = alpha[cur][tid];
        const float a1 = (tid >= 1) ? alpha[cur][tid - 1] : NEGV;
        const float a2 = allow[tid] ? alpha[cur][tid - 2] : NEGV;
        float mm = fmaxf(a0, fmaxf(a1, a2));
        float la = mm + __logf(__expf(a0 - mm) + __expf(a1 - mm) + __expf(a2 - mm));
        nv = (t < plen) ? (e + la) : a0;   // freeze past preds_len
      }
      alpha[cur ^ 1][tid] = nv;
    }
    __syncthreads();                   // reads of rowbuf[slot] & alpha[cur] done
    cur ^= 1;

    // Prefetch row t+3 into the slot we just finished reading ((t+3)%3 == t%3).
    if (t + 3 < Tn && tid < 128) {
      async_cp_b128(preds + ((size_t)(t + 3) * Bn + b) * Vn + tid * 4,
                    &rowbuf[slot][tid * 4]);
    }
    slot = (slot == 2) ? 0 : slot + 1;
  }

  if (tid == 0) {
    const float e1 = alpha[cur][2 * llen];
    int i2 = 2 * llen - 1;
    if (i2 < 0) i2 = 0;
    if (i2 > Sn - 1) i2 = Sn - 1;
    const float e2 = (llen > 0) ? alpha[cur][i2] : NEGV;
    const float mm = fmaxf(e1, e2);
    loss_out[b] = -(mm + __logf(__expf(e1 - mm) + __expf(e2 - mm)));
  }
}

// ---------------------------------------------------------------------------
// lse[b*T + t] = logsumexp_v preds[t,b,v].  One wave32 per row, 8 rows per
// 256-thread block, float4 loads, shfl-xor reductions. Streams all of preds
// through L2 (192MB L2 >> 67MB tensor) ahead of the serial kernel.
// ---------------------------------------------------------------------------
__global__ __launch_bounds__(256) void lse_kernel(const float* __restrict__ preds,
                                                  float* __restrict__ lse_out) {
  const int wave = threadIdx.x >> 5;
  const int lane = threadIdx.x & 31;
  const int row  = blockIdx.x * 8 + wave;          // row = t*Bn + b
  const float4* rp = (const float4*)(preds + (size_t)row * Vn);
  float4 c0 = rp[lane];
  float4 c1 = rp[lane + 32];
  float4 c2 = rp[lane + 64];
  float4 c3 = rp[lane + 96];

  float m = fmaxf(fmaxf(fmaxf(c0.x, c0.y), fmaxf(c0.z, c0.w)),
                  fmaxf(fmaxf(c1.x, c1.y), fmaxf(c1.z, c1.w)));
  m = fmaxf(m, fmaxf(fmaxf(c2.x, c2.y), fmaxf(c2.z, c2.w)));
  m = fmaxf(m, fmaxf(fmaxf(c3.x, c3.y), fmaxf(c3.z, c3.w)));
#pragma unroll
  for (int off = 16; off; off >>= 1) m = fmaxf(m, __shfl_xor(m, off, 32));

  float s = __expf(c0.x - m) + __expf(c0.y - m) + __expf(c0.z - m) + __expf(c0.w - m)
          + __expf(c1.x - m) + __expf(c1.y - m) + __expf(c1.z - m) + __expf(c1.w - m)
          + __expf(c2.x - m) + __expf(c2.y - m) + __expf(c2.z - m) + __expf(c2.w - m)
          + __expf(c3.x - m) + __expf(c3.y - m) + __expf(c3.z - m) + __expf(c3.w - m);
#pragma unroll
  for (int off = 16; off; off >>= 1) s += __shfl_xor(s, off, 32);

  if (lane == 0) {
    const int t = row >> 6;          // / Bn
    const int b = row & (Bn - 1);
    lse_out[b * Tn + t] = m + __logf(s);
  }
}

// ---------------------------------------------------------------------------
// Deterministic mean over B=64 per-sample losses -> d_out[0].
// ---------------------------------------------------------------------------
__global__ __launch_bounds__(32) void mean_kernel(const float* __restrict__ loss,
                                                  float* __restrict__ out) {
  float v = loss[threadIdx.x] + loss[threadIdx.x + 32];
#pragma unroll
  for (int off = 16; off; off >>= 1) v += __shfl_xor(v, off, 32);
  if (threadIdx.x == 0) out[0] = v * (1.0f / Bn);
}

extern "C" void kernel_launch(void* const* d_in, const int* in_sizes, int n_in,
                              void* d_out, int out_size, void* d_ws, size_t ws_size,
                              hipStream_t stream) {
  const float* preds  = (const float*)d_in[0];   // [T,B,V] logits
  const int*   labels = (const int*)d_in[1];     // [B*L], blank=0 excluded
  const int*   plens  = (const int*)d_in[2];     // [B]
  const int*   llens  = (const int*)d_in[3];     // [B]

  float* ws_lse  = (float*)d_ws;                 // T*B floats (128 KB)
  float* ws_loss = ws_lse + (size_t)Tn * Bn;     // B floats

  lse_kernel<<<(Tn * Bn) / 8, 256, 0, stream>>>(preds, ws_lse);
  ctc_fwd_kernel<<<Bn, 288, 0, stream>>>(preds, labels, plens, llens, ws_lse, ws_loss);
  mean_kernel<<<1, 32, 0, stream>>>(ws_loss, (float*)d_out);
}